// ScannedPredNet_37589553775142
// MI455X (gfx1250) — compile-verified
//
#include <hip/hip_runtime.h>
#include <hip/hip_bf16.h>
#include <math.h>

// Problem constants (from reference): T=128, B=256, H=1024
#define TT 128
#define BB 256
#define HH 1024
#define H2 2048
#define H3 3072

typedef __attribute__((ext_vector_type(2))) float v2f;
typedef __attribute__((ext_vector_type(4))) float v4f;
typedef __attribute__((ext_vector_type(8))) float v8f;

// ---------------------------------------------------------------------------
// Zero-init kernel for h / a_hat state at the start of each launch
// ---------------------------------------------------------------------------
__global__ __launch_bounds__(256) void zero_init_kernel(float* __restrict__ p, int n) {
    int i = blockIdx.x * blockDim.x + threadIdx.x;
    if (i < n) p[i] = 0.0f;
}

// ---------------------------------------------------------------------------
// Per-step prep: apply reset mask to h/a_hat, build E = [pos, neg], err_mag
// One block (256 threads) per batch row b; 1024 columns, 4 per thread.
// ---------------------------------------------------------------------------
__global__ __launch_bounds__(256) void prep_kernel(
    const float* __restrict__ a_target_t,       // (B,H) slice for this t
    const unsigned char* __restrict__ resets_t, // (B,) bool for this t
    float* __restrict__ h,                      // (B,H) state, masked in place
    const float* __restrict__ a_hat,            // (B,H) state (read only here)
    float* __restrict__ E,                      // (B,2H) error features out
    float* __restrict__ err_out)                // (B,) err_mags slice for this t
{
    const int b = blockIdx.x;
    const bool rst = (resets_t[b] != 0);
    __shared__ float red[256];

    float local = 0.0f;
    const int base = b * HH;
    for (int j = threadIdx.x; j < HH; j += 256) {
        float hv = rst ? 0.0f : h[base + j];
        float ah = rst ? 0.0f : a_hat[base + j];
        float at = a_target_t[base + j];
        float d   = at - ah;
        float pos = fmaxf(d, 0.0f);
        float neg = fmaxf(-d, 0.0f);
        E[b * H2 + j]      = pos;
        E[b * H2 + HH + j] = neg;
        h[base + j] = hv;            // write back masked hidden state
        local += pos + neg;
    }
    red[threadIdx.x] = local;
    __syncthreads();
    for (int s = 128; s > 0; s >>= 1) {
        if ((int)threadIdx.x < s) red[threadIdx.x] += red[threadIdx.x + s];
        __syncthreads();
    }
    if (threadIdx.x == 0) err_out[b] = red[0] * (1.0f / (float)HH);
}

// ---------------------------------------------------------------------------
// fp32 WMMA GEMM body:  C[256,N] tile = A[256,K] @ W[K,N] (+bias) (opt ReLU)
// V_WMMA_F32_16X16X4_F32, wave32.
//   - each wave computes a 32(M) x 64(N) strip: 2x4 accumulators of 16x16
//   - 8 waves per block cover M=256 (whole batch); caller picks N strip
// Fragment lane maps (ISA 7.12.2):
//   A 16x4 : m = lane&15, kk = 2*(lane>>4); v0=K(kk), v1=K(kk+1)
//   B 4x16 : n = lane&15, kk = 2*(lane>>4); v0=K(kk) row, v1=K(kk+1) row
//   C 16x16: n = lane&15, VGPR i holds row i (lanes 0-15) / row 8+i (16-31)
// ---------------------------------------------------------------------------
template <int RELU>
__device__ __forceinline__ void gemm_body(
    const float* __restrict__ A,     // (256,K) row major
    const float* __restrict__ W,     // (K,N) row major
    const float* __restrict__ bias,  // (N,) or nullptr (block-uniform)
    float* __restrict__ C,           // (256,N) row major
    int N, int K, int nBlk)
{
    const int lane  = threadIdx.x & 31;
    const int wave  = threadIdx.x >> 5;
    const int lm    = lane & 15;          // m (A) / n (B,C) within tile
    const int kk    = (lane >> 4) << 1;   // 0 or 2 : K sub-offset per lane half
    const int mBase = wave * 32;          // wave's 32-row M strip

    v8f acc[2][4];
    const v8f vzero = {};
#pragma unroll
    for (int mt = 0; mt < 2; ++mt)
#pragma unroll
        for (int nt = 0; nt < 4; ++nt) acc[mt][nt] = vzero;

    // A fragment pointers: 8-byte aligned (row base mult of 16B, kk even, k mult 4)
    const v2f* a0p = (const v2f*)__builtin_assume_aligned(
        A + (size_t)(mBase + lm) * K + kk, 8);
    const v2f* a1p = (const v2f*)__builtin_assume_aligned(
        A + (size_t)(mBase + 16 + lm) * K + kk, 8);

#pragma unroll 4
    for (int k = 0; k < K; k += 4) {
        v2f a0 = a0p[k >> 1];            // one global_load_b64
        v2f a1 = a1p[k >> 1];            // one global_load_b64

        const float* wrow = W + (size_t)(k + kk) * N + nBlk + lm;
#pragma unroll
        for (int nt = 0; nt < 4; ++nt) {
            v2f b;
            b.x = wrow[nt * 16];
            b.y = wrow[nt * 16 + N];
            acc[0][nt] = __builtin_amdgcn_wmma_f32_16x16x4_f32(
                false, a0, false, b, (short)0, acc[0][nt], false, false);
            acc[1][nt] = __builtin_amdgcn_wmma_f32_16x16x4_f32(
                false, a1, false, b, (short)0, acc[1][nt], false, false);
        }
    }

    // Epilogue: bias / relu / store
    const int rowOff = (lane >> 4) << 3;  // lanes 16-31 hold rows 8..15
#pragma unroll
    for (int mt = 0; mt < 2; ++mt) {
#pragma unroll
        for (int nt = 0; nt < 4; ++nt) {
            const int n = nBlk + nt * 16 + lm;
            float bv = bias ? bias[n] : 0.0f;
#pragma unroll
            for (int i = 0; i < 8; ++i) {
                const int m = mBase + mt * 16 + rowOff + i;
                float v = acc[mt][nt][i] + bv;
                if (RELU) v = fmaxf(v, 0.0f);
                C[(size_t)m * N + n] = v;
            }
        }
    }
}

// ---------------------------------------------------------------------------
// Fused gates GEMM: blocks [0,48) -> GI = E @ W_i + b_i   (K=2048)
//                   blocks [48,96) -> GH = h @ W_h        (K=1024)
// One launch per timestep instead of two (launch latency is on the serial
// critical path of the scan).
// ---------------------------------------------------------------------------
__global__ __launch_bounds__(256) void gates_gemm_kernel(
    const float* __restrict__ E, const float* __restrict__ h,
    const float* __restrict__ W_i, const float* __restrict__ W_h,
    const float* __restrict__ b_i,
    float* __restrict__ GI, float* __restrict__ GH)
{
    const int nb = H3 / 64;                 // 48 N-blocks per GEMM
    if ((int)blockIdx.x < nb) {
        gemm_body<0>(E, W_i, b_i, GI, H3, H2, (int)blockIdx.x * 64);
    } else {
        gemm_body<0>(h, W_h, nullptr, GH, H3, HH, ((int)blockIdx.x - nb) * 64);
    }
}

// ---------------------------------------------------------------------------
// Projection GEMM: a_hat = relu(h @ W_p + b_p)   (256x1024)@(1024x1024)
// ---------------------------------------------------------------------------
__global__ __launch_bounds__(256) void proj_gemm_kernel(
    const float* __restrict__ h, const float* __restrict__ W_p,
    const float* __restrict__ b_p, float* __restrict__ a_hat)
{
    gemm_body<1>(h, W_p, b_p, a_hat, HH, HH, (int)blockIdx.x * 64);
}

// ---------------------------------------------------------------------------
// GRU pointwise update: consumes GI (B,3H), GH (B,3H); updates h, writes r_ts
// ---------------------------------------------------------------------------
__global__ __launch_bounds__(256) void gru_pointwise_kernel(
    const float* __restrict__ GI, const float* __restrict__ GH,
    const float* __restrict__ b_hn,
    float* __restrict__ h, float* __restrict__ r_out)
{
    const int idx = blockIdx.x * blockDim.x + threadIdx.x;  // [0, B*H)
    const int b = idx >> 10;
    const int j = idx & (HH - 1);
    const float* gi = GI + (size_t)b * H3;
    const float* gh = GH + (size_t)b * H3;

    float r = 1.0f / (1.0f + expf(-(gi[j] + gh[j])));
    float z = 1.0f / (1.0f + expf(-(gi[HH + j] + gh[HH + j])));
    float n = tanhf(gi[H2 + j] + r * (gh[H2 + j] + b_hn[j]));
    float hv = h[idx];
    float nh = (1.0f - z) * n + z * hv;
    h[idx]     = nh;
    r_out[idx] = nh;
}

// ---------------------------------------------------------------------------
// kernel_launch: full T=128 scan as a chain of kernels on `stream`
// ---------------------------------------------------------------------------
extern "C" void kernel_launch(void* const* d_in, const int* in_sizes, int n_in,
                              void* d_out, int out_size, void* d_ws, size_t ws_size,
                              hipStream_t stream) {
    (void)in_sizes; (void)n_in; (void)out_size; (void)ws_size;

    const float*         a_target = (const float*)d_in[0];          // (T,B,H)
    const unsigned char* resets   = (const unsigned char*)d_in[1];  // (T,B) bool
    const float*         W_i      = (const float*)d_in[2];          // (2H,3H)
    const float*         b_i      = (const float*)d_in[3];          // (3H,)
    const float*         W_h      = (const float*)d_in[4];          // (H,3H)
    const float*         b_hn     = (const float*)d_in[5];          // (H,)
    const float*         W_p      = (const float*)d_in[6];          // (H,H)
    const float*         b_p      = (const float*)d_in[7];          // (H,)

    float* r_ts     = (float*)d_out;                 // (T,B,H)
    float* err_mags = r_ts + (size_t)TT * BB * HH;   // (T,B)

    float* ws    = (float*)d_ws;
    float* h     = ws;                        // (B,H)
    float* a_hat = h     + (size_t)BB * HH;   // (B,H)
    float* E     = a_hat + (size_t)BB * HH;   // (B,2H)
    float* GI    = E     + (size_t)BB * H2;   // (B,3H)
    float* GH    = GI    + (size_t)BB * H3;   // (B,3H)

    // h and a_hat are contiguous: zero both (2*B*H floats)
    {
        const int n = 2 * BB * HH;
        zero_init_kernel<<<(n + 255) / 256, 256, 0, stream>>>(h, n);
    }

    for (int t = 0; t < TT; ++t) {
        const float* a_t = a_target + (size_t)t * BB * HH;
        const unsigned char* rst_t = resets + (size_t)t * BB;

        // 1) reset mask + error split + err_mag reduction
        prep_kernel<<<BB, 256, 0, stream>>>(a_t, rst_t, h, a_hat, E,
                                            err_mags + (size_t)t * BB);

        // 2) fused: GI = E @ W_i + b_i  and  GH = h @ W_h
        gates_gemm_kernel<<<2 * (H3 / 64), 256, 0, stream>>>(
            E, h, W_i, W_h, b_i, GI, GH);

        // 3) GRU pointwise -> new h, write r_ts[t]
        gru_pointwise_kernel<<<(BB * HH) / 256, 256, 0, stream>>>(
            GI, GH, b_hn, h, r_ts + (size_t)t * BB * HH);

        // 4) a_hat = relu(h @ W_p + b_p)
        proj_gemm_kernel<<<HH / 64, 256, 0, stream>>>(h, W_p, b_p, a_hat);
    }
}